// GraphNorm_1451698946636
// MI455X (gfx1250) — compile-verified
//
#include <hip/hip_runtime.h>
#include <math.h>

#define FDIM 128
#define NSEG 256
#define EPS_F 1e-5f

typedef float v2f __attribute__((ext_vector_type(2)));
typedef float v8f __attribute__((ext_vector_type(8)));

__device__ __forceinline__ int lower_bound_i32(const int* __restrict__ a, int n, int v) {
    int lo = 0, hi = n;
    while (lo < hi) {
        int mid = (lo + hi) >> 1;
        if (a[mid] < v) lo = mid + 1; else hi = mid;
    }
    return lo;
}

// Issue one 16x16x4 f32 WMMA pair (sum and sum-of-squares) for 4 tiles.
__device__ __forceinline__ void wmma_accum_tiles(
    const v2f ones, const float x0[4], const float x1[4],
    v8f csum[4], v8f csq[4])
{
    #pragma unroll
    for (int t = 0; t < 4; ++t) {
        v2f b  = {x0[t], x1[t]};
        v2f bq = {x0[t] * x0[t], x1[t] * x1[t]};
        csum[t] = __builtin_amdgcn_wmma_f32_16x16x4_f32(
            false, ones, false, b,  (short)0, csum[t], false, false);
        csq[t]  = __builtin_amdgcn_wmma_f32_16x16x4_f32(
            false, ones, false, bq, (short)0, csq[t],  false, false);
    }
}

// ---------------------------------------------------------------------------
// Kernel 1: per-(segment, split) partial sums of x and x^2 per feature.
// Row reduction via V_WMMA_F32_16X16X4_F32 with A = ones(16x4):
//   D[m][n] = sum_k B[k][n]  -> every row of D holds the 4-row column sums.
// Wave layout (wave32, 8 waves/block):
//   wgrp  = wave&3       : which 4-row group this wave reduces
//   fbase = (wave>>2)*64 : which 64-feature half this wave covers (4 tiles)
// Block covers 16 rows x 128 features per step; grid.y=8 splits stride the
// 4-row units by 32 so 2048 blocks share the 500k rows.
// Tail (<4 rows) is peeled out of the hot loop so the steady state has
// unconditional b32 loads and no EXEC manipulation.
// ---------------------------------------------------------------------------
__global__ __launch_bounds__(256) void gn_partial_stats(
    const float* __restrict__ h, const int* __restrict__ seg, int n,
    float* __restrict__ wsum, float* __restrict__ wsq)
{
    __shared__ int bounds[2];
    __shared__ float psum[4][FDIM];
    __shared__ float psq[4][FDIM];

    const int g = blockIdx.x;
    if (threadIdx.x == 0) bounds[0] = lower_bound_i32(seg, n, g);
    if (threadIdx.x == 1) bounds[1] = lower_bound_i32(seg, n, g + 1);
    __syncthreads();
    const int start = bounds[0];
    const int end   = bounds[1];

    const int wave  = threadIdx.x >> 5;   // wave32
    const int lane  = threadIdx.x & 31;
    const int wgrp  = wave & 3;           // 4-row group id
    const int fbase = (wave >> 2) * 64;   // feature half (4 tiles of 16)
    const int col   = lane & 15;          // N index within a 16-wide tile
    const int half  = lane >> 4;          // which of the two B VGPR halves

    v8f csum[4] = {};
    v8f csq[4]  = {};
    const v2f ones = {1.0f, 1.0f};

    // Per-lane feature offset within the row.
    const int foff = fbase + col;
    // Rows carried by B vgpr0 / vgpr1 for this lane within a 4-row unit.
    const int offA = half ? 2 : 0;
    const int offB = half ? 3 : 1;

    int r0 = start + 4 * (blockIdx.y * 4 + wgrp);   // first row of my unit
    // ---- hot loop: full 4-row units only, unconditional loads ----
    for (; r0 + 4 <= end; r0 += 128) {              // stride = 32 units
        const float* pa = h + (size_t)(r0 + offA) * FDIM + foff;
        const float* pb = h + (size_t)(r0 + offB) * FDIM + foff;
        float x0[4], x1[4];
        #pragma unroll
        for (int t = 0; t < 4; ++t) { x0[t] = pa[t * 16]; x1[t] = pb[t * 16]; }
        wmma_accum_tiles(ones, x0, x1, csum, csq);
    }
    // ---- peeled tail: at most one partial unit (1..3 rows) per segment ----
    if (r0 < end) {
        const int ra = r0 + offA;
        const int rb = r0 + offB;
        const float* pa = h + (size_t)ra * FDIM + foff;
        const float* pb = h + (size_t)rb * FDIM + foff;
        const bool va = ra < end;
        const bool vb = rb < end;
        float x0[4], x1[4];
        #pragma unroll
        for (int t = 0; t < 4; ++t) {
            x0[t] = va ? pa[t * 16] : 0.0f;
            x1[t] = vb ? pb[t * 16] : 0.0f;
        }
        wmma_accum_tiles(ones, x0, x1, csum, csq);
    }

    // C VGPR0 on lanes 0..15 = D row M=0, N=lane -> the per-feature sums.
    if (lane < 16) {
        #pragma unroll
        for (int t = 0; t < 4; ++t) {
            psum[wgrp][fbase + t * 16 + lane] = csum[t][0];
            psq [wgrp][fbase + t * 16 + lane] = csq[t][0];
        }
    }
    __syncthreads();

    if (threadIdx.x < FDIM) {
        const int f = threadIdx.x;
        const float s = psum[0][f] + psum[1][f] + psum[2][f] + psum[3][f];
        const float q = psq[0][f]  + psq[1][f]  + psq[2][f]  + psq[3][f];
        atomicAdd(&wsum[g * FDIM + f], s);
        atomicAdd(&wsq [g * FDIM + f], q);
    }
}

// ---------------------------------------------------------------------------
// Kernel 2: finalize stats -> fused affine tables.
//   gscale = gamma / (sqrt(var) + eps),  gbias = beta - mean * gscale
// ---------------------------------------------------------------------------
__global__ __launch_bounds__(128) void gn_finalize(
    const int* __restrict__ seg, int n,
    const float* __restrict__ gamma, const float* __restrict__ beta,
    const float* __restrict__ wsum, const float* __restrict__ wsq,
    float* __restrict__ gsc, float* __restrict__ gbi)
{
    __shared__ int bounds[2];
    const int g = blockIdx.x;
    if (threadIdx.x == 0) bounds[0] = lower_bound_i32(seg, n, g);
    if (threadIdx.x == 1) bounds[1] = lower_bound_i32(seg, n, g + 1);
    __syncthreads();
    const float cnt = (float)(bounds[1] - bounds[0]);

    const int f   = threadIdx.x;
    const int idx = g * FDIM + f;
    const float s = wsum[idx];
    const float q = wsq[idx];
    const float mean = s / cnt;
    const float var  = fmaf(-s, mean, q) / (cnt - 1.0f);  // (q - s^2/n)/(n-1)
    const float scale = 1.0f / (sqrtf(var) + EPS_F);
    const float G = gamma[f] * scale;
    gsc[idx] = G;
    gbi[idx] = fmaf(-mean, G, beta[f]);
}

// ---------------------------------------------------------------------------
// Kernel 3: streaming apply, one FMA per element, b128 loads/stores.
// ---------------------------------------------------------------------------
__global__ __launch_bounds__(256) void gn_apply(
    const float* __restrict__ h, const int* __restrict__ seg,
    const float* __restrict__ gsc, const float* __restrict__ gbi,
    float* __restrict__ out, int n)
{
    const long long idx  = (long long)blockIdx.x * blockDim.x + threadIdx.x; // one float4
    const long long nvec = (long long)n * (FDIM / 4);
    if (idx >= nvec) return;
    const int row = (int)(idx >> 5);           // FDIM/4 == 32 float4 per row
    const int c4  = ((int)idx & 31) << 2;
    const int g   = seg[row];

    const float4 x = ((const float4*)h)[idx];
    const float4 a = *(const float4*)(gsc + (size_t)g * FDIM + c4);
    const float4 b = *(const float4*)(gbi + (size_t)g * FDIM + c4);
    float4 o;
    o.x = fmaf(x.x, a.x, b.x);
    o.y = fmaf(x.y, a.y, b.y);
    o.z = fmaf(x.z, a.z, b.z);
    o.w = fmaf(x.w, a.w, b.w);
    ((float4*)out)[idx] = o;
}

extern "C" void kernel_launch(void* const* d_in, const int* in_sizes, int n_in,
                              void* d_out, int out_size, void* d_ws, size_t ws_size,
                              hipStream_t stream) {
    const float* h     = (const float*)d_in[0];
    const int*   seg   = (const int*)d_in[1];
    const float* gamma = (const float*)d_in[2];
    const float* beta  = (const float*)d_in[3];
    const int n = in_sizes[1];            // N rows (segment_ids length)
    float* out = (float*)d_out;

    float* wsum = (float*)d_ws;           // [256][128]
    float* wsq  = wsum + NSEG * FDIM;     // [256][128]
    float* gsc  = wsq  + NSEG * FDIM;     // [256][128]
    float* gbi  = gsc  + NSEG * FDIM;     // [256][128]  -> 512 KB total

    hipMemsetAsync(d_ws, 0, sizeof(float) * NSEG * FDIM * 2, stream);

    dim3 grid1(NSEG, 8);
    gn_partial_stats<<<grid1, 256, 0, stream>>>(h, seg, n, wsum, wsq);
    gn_finalize<<<NSEG, 128, 0, stream>>>(seg, n, gamma, beta, wsum, wsq, gsc, gbi);

    const long long nvec = (long long)n * (FDIM / 4);
    const int blocks = (int)((nvec + 255) / 256);
    gn_apply<<<blocks, 256, 0, stream>>>(h, seg, gsc, gbi, out, n);
}